// Fssn_layers_18391049962175
// MI455X (gfx1250) — compile-verified
//
#include <hip/hip_runtime.h>

// GAT-like heterogeneous aggregation for MI455X (gfx1250), wave32.
//
// Structure (memory-bound; batch_features [16MB] lives in the 192MB L2):
//   1) init:    d_out[16M f32] = -inf
//   2) scatter: one wave per batch row. V_WMMA_F32_16X16X4_F32 computes all
//      16 (filter,head) x 16-feature partial results per slice; lanes scatter
//      their 8 D-VGPRs via f32 max-atomics (global_atomic_max_num_f32).
//   3) finalize: out = leaky_relu(out + batch_features), fused in-place.

#define ALPHA    0.2f
#define N_FEAT   64
#define N_HEADS  4
#define OUT_PER_NODE (N_HEADS * N_FEAT)   // 256

typedef __attribute__((ext_vector_type(2))) float v2f;
typedef __attribute__((ext_vector_type(8))) float v8f;

__global__ __launch_bounds__(256) void gat_init_kernel(float* __restrict__ out) {
    size_t idx = (size_t)blockIdx.x * 256 + threadIdx.x;
    out[idx] = -__builtin_inff();
}

// One wave32 per batch row. WMMA mapping:
//   K (=4)  : the 4 nodes of the batch row (j = 0..3)
//   M (=16) : m = i*4 + a  (filter i = target slot, head a)
//   N (=16) : one 16-wide feature slice (4 slices cover F=64)
//   A[m, j] = (j == i) ? 0 : attw[a, j<i ? j : j-1]   (constant per kernel)
//   B[j, n] = feat[batch[b, j], fb + n]
//   D[m, n] = att_feat for (target i, head a, feature fb+n) -> atomic fmax.
__global__ __launch_bounds__(256) void gat_scatter_wmma_kernel(
    const int*   __restrict__ batch,   // [B, 4]
    const float* __restrict__ feat,    // [N_NODES, 64]
    const float* __restrict__ attw,    // [4, 3]
    float*       __restrict__ out) {   // [N_NODES, 256] accumulator (-inf init)
    const int lane = threadIdx.x & 31;
    const int wave = threadIdx.x >> 5;
    const int b    = blockIdx.x * 8 + wave;        // grid sized exactly: no partial waves

    // ---- A matrix (16x4 f32): lanes 0-15 hold M=lane, K = {0,1}; lanes 16-31 K = {2,3}
    const int m     = lane & 15;
    const int i     = m >> 2;        // filter / target slot
    const int a     = m & 3;         // head
    const int hi    = lane >> 4;
    const int kbase = hi << 1;

    float a0, a1;
    {
        const int k0 = kbase, k1 = kbase + 1;
        a0 = (k0 == i) ? 0.0f : attw[a * 3 + (k0 < i ? k0 : k0 - 1)];
        a1 = (k1 == i) ? 0.0f : attw[a * 3 + (k1 < i ? k1 : k1 - 1)];
    }
    v2f A; A.x = a0; A.y = a1;

    // ---- node ids for this batch row (uniform across the wave)
    const int4 ids = ((const int4*)batch)[b];
    // B rows this lane supplies (K = kbase, kbase+1); also the scatter targets
    // for its D registers (m = p + 8*hi -> i = 2*hi + (p>=4)): same two ids.
    const int nk0 = hi ? ids.z : ids.x;
    const int nk1 = hi ? ids.w : ids.y;

    const float* __restrict__ rowk0 = feat + (size_t)nk0 * N_FEAT;
    const float* __restrict__ rowk1 = feat + (size_t)nk1 * N_FEAT;
    const int col = lane & 15;       // N coordinate

#pragma unroll
    for (int s = 0; s < 4; ++s) {
        const int fb = s * 16;
        // B matrix (4x16 f32): lanes 0-15 N=lane K={0,1}; lanes 16-31 N=lane-16 K={2,3}
        v2f B; B.x = rowk0[fb + col]; B.y = rowk1[fb + col];
        v8f c = {};
        // (neg_a, A, neg_b, B, c_mod, C, reuse_a, reuse_b)
        v8f d = __builtin_amdgcn_wmma_f32_16x16x4_f32(
            false, A, false, B, (short)0, c, false, false);

        // D layout: VGPR p holds (M = p + 8*hi, N = col); target node = ids[M>>2]
#pragma unroll
        for (int p = 0; p < 8; ++p) {
            const int node = (p < 4) ? nk0 : nk1;   // ids[2*hi + (p>=4)]
            const int aa   = p & 3;                  // head of row M = p + 8*hi
            float* addr = out + (size_t)node * OUT_PER_NODE + aa * N_FEAT + fb + col;
            (void)__hip_atomic_fetch_max(addr, d[p], __ATOMIC_RELAXED,
                                         __HIP_MEMORY_SCOPE_AGENT);
        }
    }
}

__global__ __launch_bounds__(256) void gat_finalize_kernel(
    const float* __restrict__ feat, float* __restrict__ out) {
    size_t idx = (size_t)blockIdx.x * 256 + threadIdx.x;
    const size_t n = idx >> 8;          // / OUT_PER_NODE
    const int    f = (int)(idx & 63);   // feature within head
    float v = out[idx] + feat[n * N_FEAT + f];
    out[idx] = (v >= 0.0f) ? v : ALPHA * v;
}

extern "C" void kernel_launch(void* const* d_in, const int* in_sizes, int n_in,
                              void* d_out, int out_size, void* d_ws, size_t ws_size,
                              hipStream_t stream) {
    const int*   batch = (const int*)d_in[0];    // [B, 4] int32
    const float* feat  = (const float*)d_in[1];  // [N_NODES, 64] f32
    const float* attw  = (const float*)d_in[2];  // [4, 3] f32
    float*       out   = (float*)d_out;          // [N_NODES, 256] f32

    const int rows      = in_sizes[0] / 4;       // 131072 batch rows
    const int out_elems = out_size;              // 16,777,216

    gat_init_kernel<<<out_elems / 256, 256, 0, stream>>>(out);
    gat_scatter_wmma_kernel<<<rows / 8, 256, 0, stream>>>(batch, feat, attw, out);
    gat_finalize_kernel<<<out_elems / 256, 256, 0, stream>>>(feat, out);
}